// GIN_22668837388504
// MI455X (gfx1250) — compile-verified
//
#include <hip/hip_runtime.h>
#include <cstdint>
#include <cstddef>

#define N_NODES 100000
#define IN_CH   128
#define HID     30
#define HPAD    32
#define MTILES  (N_NODES / 16)   // 6250, exact

typedef __attribute__((ext_vector_type(2))) float v2f;
typedef __attribute__((ext_vector_type(8))) float v8f;

// ---------------- zero workspace region ----------------
__global__ void gin_zero_kernel(float4* __restrict__ p, int n4) {
  int i = blockIdx.x * blockDim.x + threadIdx.x;
  if (i < n4) p[i] = make_float4(0.f, 0.f, 0.f, 0.f);
}

// ---------------- pack W[kReal,30] -> Wt[32][kPad], transposed + zero-padded ----------------
// Wt[n*kPad + k] = W[k*30 + n]  (zeros outside). Makes B fragments a single aligned b64 load.
__global__ void gin_pack_w(const float* __restrict__ W, int kReal, int kPad,
                           float* __restrict__ Wt) {
  int i = blockIdx.x * blockDim.x + threadIdx.x;
  if (i >= HPAD * kPad) return;
  const int n = i / kPad;
  const int k = i % kPad;
  Wt[i] = (k < kReal && n < HID) ? W[(size_t)k * HID + n] : 0.f;
}

// ---------------- P[N,32] = A[N,KPAD] @ Wt^T  (fp32 WMMA 16x16x4) ----------------
// One wave per block; one 16-node M tile; two 16-wide N tiles. No guards, no divergence.
template <int KPAD>
__global__ __launch_bounds__(32) void gin_gemm_wmma(const float* __restrict__ A,
                                                    const float* __restrict__ Wt,
                                                    float* __restrict__ P) {
  const int lane = threadIdx.x;
  const int m    = lane & 15;     // M row for A frag; N col for C/D frag
  const int hi   = lane >> 4;     // half-wave select
  const int base = blockIdx.x * 16;

  const float* arow = A + (size_t)(base + m) * KPAD;
  const float* w0   = Wt + (size_t)m * KPAD;          // column m
  const float* w1   = Wt + (size_t)(16 + m) * KPAD;   // column 16+m (zero-padded if >= 30)

  v8f c0 = {0.f,0.f,0.f,0.f,0.f,0.f,0.f,0.f};
  v8f c1 = c0;

  #pragma unroll
  for (int kk = 0; kk < KPAD; kk += 4) {
    const int ka = kk + hi * 2;                 // lanes 0-15: K=kk,kk+1 ; lanes 16-31: K=kk+2,kk+3
    v2f a  = *(const v2f*)(arow + ka);
    v2f b0 = *(const v2f*)(w0 + ka);
    v2f b1 = *(const v2f*)(w1 + ka);
    c0 = __builtin_amdgcn_wmma_f32_16x16x4_f32(false, a, false, b0, (short)0, c0, false, false);
    c1 = __builtin_amdgcn_wmma_f32_16x16x4_f32(false, a, false, b1, (short)0, c1, false, false);
  }

  #pragma unroll
  for (int r = 0; r < 8; ++r) {
    const int node = base + r + hi * 8;
    P[(size_t)node * HPAD + m]      = c0[r];
    P[(size_t)node * HPAD + 16 + m] = c1[r];
  }
}

// ---------------- AGG[dst] += P[src] over edges (32 ch, float4 gather, f32 atomics) ------------
__global__ void gin_scatter_kernel(const long long* __restrict__ ei, int nE,
                                   const float* __restrict__ P, float* __restrict__ AGG) {
  int e = blockIdx.x * blockDim.x + threadIdx.x;
  if (e >= nE) return;
  const int s = (int)ei[e];                 // src row
  const int d = (int)ei[(size_t)nE + e];    // dst row
  const float4* pv = (const float4*)(P + (size_t)s * HPAD);
  float* out = AGG + (size_t)d * HPAD;
  #pragma unroll
  for (int i = 0; i < 8; ++i) {
    float4 v = pv[i];
    atomicAdd(out + 4 * i + 0, v.x);
    atomicAdd(out + 4 * i + 1, v.y);
    atomicAdd(out + 4 * i + 2, v.z);
    atomicAdd(out + 4 * i + 3, v.w);
  }
}

// ---------------- H = ReLU( ReLU(P + AGG + ba) @ Wbt^T + bb )  (fused, WMMA f32) ---------------
__global__ __launch_bounds__(32) void gin_mlp_wmma(const float* __restrict__ P,
                                                   const float* __restrict__ AGG,
                                                   const float* __restrict__ ba,
                                                   const float* __restrict__ Wbt, // packed [32][32]
                                                   const float* __restrict__ bb,
                                                   float* __restrict__ H) {
  __shared__ float tile[16 * 34];            // stride 34 -> 8B-aligned v2f rows, conflict-free
  const int lane = threadIdx.x;
  const int m    = lane & 15;
  const int hi   = lane >> 4;
  const int base = blockIdx.x * 16;

  // phase 1: hidden = ReLU(P + AGG + ba), staged in LDS (A-matrix of second GEMM)
  const float bcol = (lane < HID) ? ba[lane] : 0.f;
  #pragma unroll 4
  for (int r = 0; r < 16; ++r) {
    const size_t g = (size_t)(base + r) * HPAD + lane;
    tile[r * 34 + lane] = fmaxf(P[g] + AGG[g] + bcol, 0.f);
  }
  __syncthreads();

  // phase 2: out = hidden @ Wb + bb  (K = 32 padded; pad rows/cols are zero)
  const float* w0 = Wbt + (size_t)m * HPAD;
  const float* w1 = Wbt + (size_t)(16 + m) * HPAD;
  v8f c0, c1;
  const float bb0 = bb[m];
  const float bb1 = (16 + m < HID) ? bb[16 + m] : 0.f;
  #pragma unroll
  for (int r = 0; r < 8; ++r) { c0[r] = bb0; c1[r] = bb1; }

  #pragma unroll
  for (int kk = 0; kk < HPAD; kk += 4) {
    const int ka = kk + hi * 2;
    v2f a  = *(const v2f*)(tile + m * 34 + ka);
    v2f b0 = *(const v2f*)(w0 + ka);
    v2f b1 = *(const v2f*)(w1 + ka);
    c0 = __builtin_amdgcn_wmma_f32_16x16x4_f32(false, a, false, b0, (short)0, c0, false, false);
    c1 = __builtin_amdgcn_wmma_f32_16x16x4_f32(false, a, false, b1, (short)0, c1, false, false);
  }

  // phase 3: outer ReLU, store padded (cols 30,31 end up exactly 0)
  #pragma unroll
  for (int r = 0; r < 8; ++r) {
    const int node = base + r + hi * 8;
    H[(size_t)node * HPAD + m]      = fmaxf(c0[r], 0.f);
    H[(size_t)node * HPAD + 16 + m] = fmaxf(c1[r], 0.f);
  }
}

// ---------------- log_softmax over 30 real channels ----------------
__global__ void gin_logsoftmax_kernel(const float* __restrict__ H, float* __restrict__ out) {
  int node = blockIdx.x * blockDim.x + threadIdx.x;
  if (node >= N_NODES) return;
  const float* h = H + (size_t)node * HPAD;
  float mx = h[0];
  #pragma unroll
  for (int c = 1; c < HID; ++c) mx = fmaxf(mx, h[c]);
  float s = 0.f;
  #pragma unroll
  for (int c = 0; c < HID; ++c) s += __expf(h[c] - mx);
  const float ls = __logf(s);
  #pragma unroll
  for (int c = 0; c < HID; ++c) out[(size_t)node * HID + c] = h[c] - mx - ls;
}

extern "C" void kernel_launch(void* const* d_in, const int* in_sizes, int n_in,
                              void* d_out, int out_size, void* d_ws, size_t ws_size,
                              hipStream_t stream) {
  const float*     x  = (const float*)d_in[0];
  const long long* ei = (const long long*)d_in[1];
  const float* w1a = (const float*)d_in[2];  const float* b1a = (const float*)d_in[3];
  const float* w1b = (const float*)d_in[4];  const float* b1b = (const float*)d_in[5];
  const float* w2a = (const float*)d_in[6];  const float* b2a = (const float*)d_in[7];
  const float* w2b = (const float*)d_in[8];  const float* b2b = (const float*)d_in[9];
  const float* w3a = (const float*)d_in[10]; const float* b3a = (const float*)d_in[11];
  const float* w3b = (const float*)d_in[12]; const float* b3b = (const float*)d_in[13];
  float* out = (float*)d_out;

  const int nE = in_sizes[1] / 2;

  const size_t featElems = (size_t)N_NODES * HPAD;          // 3.2M floats = 12.8MB each
  float* P   = (float*)d_ws;
  float* AGG = P + featElems;
  float* H   = AGG + featElems;
  float* wts = H + featElems;                                // packed weights, ~36KB total
  float* wt1a = wts;                 // [32][128]
  float* wt1b = wt1a + HPAD * IN_CH; // [32][32]
  float* wt2a = wt1b + HPAD * HPAD;
  float* wt2b = wt2a + HPAD * HPAD;
  float* wt3a = wt2b + HPAD * HPAD;
  float* wt3b = wt3a + HPAD * HPAD;

  // pack all weights (tiny; runs every call -> deterministic)
  gin_pack_w<<<(HPAD * IN_CH + 255) / 256, 256, 0, stream>>>(w1a, IN_CH, IN_CH, wt1a);
  gin_pack_w<<<(HPAD * HPAD  + 255) / 256, 256, 0, stream>>>(w1b, HID, HPAD, wt1b);
  gin_pack_w<<<(HPAD * HPAD  + 255) / 256, 256, 0, stream>>>(w2a, HID, HPAD, wt2a);
  gin_pack_w<<<(HPAD * HPAD  + 255) / 256, 256, 0, stream>>>(w2b, HID, HPAD, wt2b);
  gin_pack_w<<<(HPAD * HPAD  + 255) / 256, 256, 0, stream>>>(w3a, HID, HPAD, wt3a);
  gin_pack_w<<<(HPAD * HPAD  + 255) / 256, 256, 0, stream>>>(w3b, HID, HPAD, wt3b);

  const int zeroBlocks    = (int)((featElems / 4 + 255) / 256);
  const int scatterBlocks = (nE + 255) / 256;
  const int lsmBlocks     = (N_NODES + 255) / 256;

  const float* Wta[3] = { wt1a, wt2a, wt3a };
  const float* Wtb[3] = { wt1b, wt2b, wt3b };
  const float* Ba[3]  = { b1a, b2a, b3a };
  const float* Bb[3]  = { b1b, b2b, b3b };

  for (int layer = 0; layer < 3; ++layer) {
    // P = input @ Wa  (linearity: project BEFORE aggregating; layer 0 input is x [128-wide])
    if (layer == 0)
      gin_gemm_wmma<IN_CH><<<MTILES, 32, 0, stream>>>(x, Wta[0], P);
    else
      gin_gemm_wmma<HPAD><<<MTILES, 32, 0, stream>>>(H, Wta[layer], P);

    // AGG = 0 ; AGG[dst] += P[src]
    gin_zero_kernel<<<zeroBlocks, 256, 0, stream>>>((float4*)AGG, (int)(featElems / 4));
    gin_scatter_kernel<<<scatterBlocks, 256, 0, stream>>>(ei, nE, P, AGG);

    // H = ReLU( ReLU(P + AGG + ba) @ Wb + bb )
    gin_mlp_wmma<<<MTILES, 32, 0, stream>>>(P, AGG, Ba[layer], Wtb[layer], Bb[layer], H);
  }

  gin_logsoftmax_kernel<<<lsmBlocks, 256, 0, stream>>>(H, out);
}